// ConditionalMambaResidualBlock1D_26499948216666
// MI455X (gfx1250) — compile-verified
//
#include <hip/hip_runtime.h>
#include <hip/hip_bf16.h>

// ---------------------------------------------------------------------------
// ConditionalMambaResidualBlock1D for MI455X (gfx1250, wave32, WMMA).
// All dense GEMMs run through V_WMMA_F32_16X16X4_F32 (fp32 in/out, matches
// the fp32 reference numerics). One generic tiled GEMM kernel is reused for
// conv(im2col), FiLM, in_proj, x_proj, dt_proj, out_proj, fc1, fc2, res-1x1,
// each with a fused epilogue. B panel is staged fragment-major in LDS so each
// WMMA B operand is a single contiguous ds_load_b64 (no VALU repacking).
// ---------------------------------------------------------------------------

typedef float v2f __attribute__((ext_vector_type(2)));
typedef float v8f __attribute__((ext_vector_type(8)));

#define Bsz   32
#define CIN   256
#define COUT  512
#define Lseq  256
#define DIN   1024
#define DSTATE 16
#define DTR   32
#define MLPH  2048
#define NG    8
#define Mrows (Bsz * Lseq)   // 8192

// GEMM tiling: block tile 64(M) x 128(N); 8 waves = 4(M) x 2(N);
// per-wave tile 16 x 64 (4 accumulators) -> 4 WMMA per A fragment.
#define TM  64
#define TNB 128
#define KC  32

enum { A_NORMAL = 0, A_CONV = 1 };
enum { EPI_STORE = 0, EPI_BIAS = 1, EPI_BIAS_SOFTPLUS = 2, EPI_BIAS_GELU = 3,
       EPI_ADD = 4, EPI_BIAS_ADD = 5, EPI_BIAS_ADD_TRANS = 6 };

__device__ __forceinline__ float softplusf(float x) {
    return (x > 20.f) ? x : log1pf(__expf(x));
}
__device__ __forceinline__ float mishf(float x) {
    return x * tanhf(softplusf(x));
}
__device__ __forceinline__ float siluf(float x) {
    return x / (1.f + __expf(-x));
}
__device__ __forceinline__ float geluf(float x) {  // exact erf gelu
    return 0.5f * x * (1.f + erff(x * 0.70710678118654752f));
}

template<int EPI>
__device__ __forceinline__ void epi_store(float x, int row, int col, int N,
                                          const float* __restrict__ bias,
                                          const float* __restrict__ R,
                                          float* __restrict__ C) {
    if (EPI == EPI_BIAS || EPI == EPI_BIAS_SOFTPLUS || EPI == EPI_BIAS_GELU ||
        EPI == EPI_BIAS_ADD || EPI == EPI_BIAS_ADD_TRANS)
        x += bias[col];
    if (EPI == EPI_BIAS_SOFTPLUS) x = softplusf(x);
    if (EPI == EPI_BIAS_GELU)     x = geluf(x);
    if (EPI == EPI_ADD || EPI == EPI_BIAS_ADD || EPI == EPI_BIAS_ADD_TRANS)
        x += R[(size_t)row * N + col];
    if (EPI == EPI_BIAS_ADD_TRANS) {
        // final output: [B, COUT, L] from row=(b*L+l), col=co
        int b = row >> 8, l = row & 255;
        C[((size_t)b * N + col) * Lseq + l] = x;
    } else {
        C[(size_t)row * N + col] = x;
    }
}

// C[M,N] = A[M,K] @ Bm[K,N]  (+ epilogue). A_CONV: A is xT [B,L,CIN] and the
// logical A row (b*L+l) is the contiguous slice xT[b, l-1 : l+2, :] (768
// floats) with zero padding at l==0 / l==L-1. K must be a multiple of KC;
// M and N may be ragged (guarded).
template<int AMODE, int EPI>
__global__ __launch_bounds__(256)
void gemm_wmma_f32(const float* __restrict__ A, int lda,
                   const float* __restrict__ Bm,
                   const float* __restrict__ bias,
                   const float* __restrict__ R,
                   float* __restrict__ C,
                   int M, int N, int K) {
    __shared__ float As[TM][KC + 4];    // row-major panel; frag = contiguous k pair
    __shared__ float Bs[TNB][KC + 2];   // FRAGMENT-MAJOR (transposed) panel:
                                        // Bs[col][k] = Bm[kt+k][bn+col]
    const int tid  = threadIdx.x;
    const int bm   = blockIdx.x * TM;
    const int bn   = blockIdx.y * TNB;
    const int wave = tid >> 5;
    const int lane = tid & 31;
    const int wm   = (wave & 3) * 16;   // 4 waves along M
    const int wn   = (wave >> 2) * 64;  // 2 waves along N, 4 subtiles each
    const int half = lane >> 4;
    const int lid  = lane & 15;

    v8f acc0 = {}, acc1 = {}, acc2 = {}, acc3 = {};

    for (int kt = 0; kt < K; kt += KC) {
        // ---- stage A panel [TM x KC] (row-major, float4 loads/stores) ----
        {
            int r  = tid >> 3;          // 0..31
            int c4 = (tid & 7) * 4;     // 0..28
            #pragma unroll
            for (int rr = 0; rr < 2; ++rr) {
                int row  = r + rr * 32;
                int grow = bm + row;
                float4 v = make_float4(0.f, 0.f, 0.f, 0.f);
                if (AMODE == A_NORMAL) {
                    if (grow < M)
                        v = *(const float4*)(A + (size_t)grow * lda + kt + c4);
                } else {
                    int l = grow & 255, b = grow >> 8;
                    int j = kt + c4;                   // 0..767; float4 never
                    int idx = (l - 1) * CIN + j;      //   crosses a 256 block
                    if (idx >= 0 && idx < Lseq * CIN)
                        v = *(const float4*)(A + (size_t)b * Lseq * CIN + idx);
                }
                As[row][c4 + 0] = v.x; As[row][c4 + 1] = v.y;
                As[row][c4 + 2] = v.z; As[row][c4 + 3] = v.w;
            }
        }
        // ---- stage B panel transposed: Bs[col][k] ----
        // Each task: one k-pair (2 rows) x 4 cols -> 4 float2 LDS stores.
        {
            #pragma unroll
            for (int it = 0; it < 2; ++it) {
                int idx = tid + it * 256;        // 0..511
                int kp  = idx >> 5;              // 0..15 k-pair
                int c4  = (idx & 31) * 4;        // 0..124
                int gc  = bn + c4;
                float4 v0 = make_float4(0.f, 0.f, 0.f, 0.f);
                float4 v1 = make_float4(0.f, 0.f, 0.f, 0.f);
                if (gc < N) {
                    v0 = *(const float4*)(Bm + (size_t)(kt + 2 * kp)     * N + gc);
                    v1 = *(const float4*)(Bm + (size_t)(kt + 2 * kp + 1) * N + gc);
                }
                *(float2*)&Bs[c4 + 0][2 * kp] = make_float2(v0.x, v1.x);
                *(float2*)&Bs[c4 + 1][2 * kp] = make_float2(v0.y, v1.y);
                *(float2*)&Bs[c4 + 2][2 * kp] = make_float2(v0.z, v1.z);
                *(float2*)&Bs[c4 + 3][2 * kp] = make_float2(v0.w, v1.w);
            }
        }
        __syncthreads();

        #pragma unroll
        for (int kk = 0; kk < KC; kk += 4) {
            // A frag 16x4 f32: v0 holds K{kk+2h}, v1 holds K{kk+2h+1}
            v2f a = *(const v2f*)&As[wm + lid][kk + 2 * half];
            // B frags: contiguous k-pair per lane -> single ds_load_b64 each
            v2f b0 = *(const v2f*)&Bs[wn + lid     ][kk + 2 * half];
            v2f b1 = *(const v2f*)&Bs[wn + 16 + lid][kk + 2 * half];
            v2f b2 = *(const v2f*)&Bs[wn + 32 + lid][kk + 2 * half];
            v2f b3 = *(const v2f*)&Bs[wn + 48 + lid][kk + 2 * half];
            acc0 = __builtin_amdgcn_wmma_f32_16x16x4_f32(
                       false, a, false, b0, (short)0, acc0, false, false);
            acc1 = __builtin_amdgcn_wmma_f32_16x16x4_f32(
                       false, a, false, b1, (short)0, acc1, false, false);
            acc2 = __builtin_amdgcn_wmma_f32_16x16x4_f32(
                       false, a, false, b2, (short)0, acc2, false, false);
            acc3 = __builtin_amdgcn_wmma_f32_16x16x4_f32(
                       false, a, false, b3, (short)0, acc3, false, false);
        }
        __syncthreads();
    }

    // ---- epilogue: C/D layout: VGPR v, lanes0-15 -> M=v, lanes16-31 -> M=v+8
    #pragma unroll
    for (int v = 0; v < 8; ++v) {
        int row = bm + wm + half * 8 + v;
        if (row < M) {
            int col = bn + wn + lid;
            if (col      < N) epi_store<EPI>(acc0[v], row, col,      N, bias, R, C);
            if (col + 16 < N) epi_store<EPI>(acc1[v], row, col + 16, N, bias, R, C);
            if (col + 32 < N) epi_store<EPI>(acc2[v], row, col + 32, N, bias, R, C);
            if (col + 48 < N) epi_store<EPI>(acc3[v], row, col + 48, N, bias, R, C);
        }
    }
}

// ---------------------------------------------------------------------------
// small prep / elementwise kernels
// ---------------------------------------------------------------------------
__global__ void transpose_x_k(const float* __restrict__ x, float* __restrict__ xT) {
    int idx = blockIdx.x * 256 + threadIdx.x;            // B*L*CIN
    int c = idx & 255, l = (idx >> 8) & 255, b = idx >> 16;
    xT[idx] = x[((size_t)b * CIN + c) * Lseq + l];
}
__global__ void prep_wc_k(const float* __restrict__ conv_w, float* __restrict__ Wc) {
    int idx = blockIdx.x * 256 + threadIdx.x;            // 768*512
    int co = idx & 511, kc = idx >> 9;                   // kc = k*256+ci
    int k = kc >> 8, ci = kc & 255;
    Wc[idx] = conv_w[((size_t)co * CIN + ci) * 3 + k];
}
__global__ void prep_reswt_k(const float* __restrict__ res_w, float* __restrict__ W) {
    int idx = blockIdx.x * 256 + threadIdx.x;            // 256*512
    int o = idx & 511, c = idx >> 9;
    W[idx] = res_w[(size_t)o * CIN + c];
}
__global__ void mish_k(const float* __restrict__ in, float* __restrict__ out, int n) {
    int idx = blockIdx.x * 256 + threadIdx.x;
    if (idx < n) out[idx] = mishf(in[idx]);
}

// GroupNorm(8) + affine + mish + FiLM, in place on t [B, L, COUT]
__global__ __launch_bounds__(256)
void gn_film_k(float* __restrict__ t, const float* __restrict__ gn_w,
               const float* __restrict__ gn_b, const float* __restrict__ embed) {
    int b = blockIdx.x >> 3, g = blockIdx.x & 7;
    int c0 = g * 64;
    const size_t base = (size_t)b * Lseq * COUT;
    float s = 0.f, ss = 0.f;
    for (int i = threadIdx.x; i < 64 * Lseq; i += 256) {
        int l = i >> 6, c = c0 + (i & 63);
        float v = t[base + (size_t)l * COUT + c];
        s += v; ss += v * v;
    }
    __shared__ float r0[256], r1[256];
    r0[threadIdx.x] = s; r1[threadIdx.x] = ss; __syncthreads();
    for (int off = 128; off > 0; off >>= 1) {
        if (threadIdx.x < off) { r0[threadIdx.x] += r0[threadIdx.x + off];
                                 r1[threadIdx.x] += r1[threadIdx.x + off]; }
        __syncthreads();
    }
    float mu   = r0[0] * (1.f / 16384.f);
    float var  = r1[0] * (1.f / 16384.f) - mu * mu;
    float rstd = rsqrtf(var + 1e-5f);
    for (int i = threadIdx.x; i < 64 * Lseq; i += 256) {
        int l = i >> 6, c = c0 + (i & 63);
        size_t idx = base + (size_t)l * COUT + c;
        float v = (t[idx] - mu) * rstd * gn_w[c] + gn_b[c];
        v = mishf(v);
        float sc = embed[(size_t)b * (2 * COUT) + c];
        float bi = embed[(size_t)b * (2 * COUT) + COUT + c];
        t[idx] = sc * v + bi;
    }
}

// LayerNorm over last dim (512), one block per row
__global__ __launch_bounds__(256)
void layernorm_k(const float* __restrict__ in, const float* __restrict__ w,
                 const float* __restrict__ b, float* __restrict__ out) {
    const float* p = in + (size_t)blockIdx.x * COUT;
    float s = 0.f, ss = 0.f;
    for (int c = threadIdx.x; c < COUT; c += 256) { float v = p[c]; s += v; ss += v * v; }
    __shared__ float r0[256], r1[256];
    r0[threadIdx.x] = s; r1[threadIdx.x] = ss; __syncthreads();
    for (int off = 128; off > 0; off >>= 1) {
        if (threadIdx.x < off) { r0[threadIdx.x] += r0[threadIdx.x + off];
                                 r1[threadIdx.x] += r1[threadIdx.x + off]; }
        __syncthreads();
    }
    float mu   = r0[0] * (1.f / COUT);
    float rstd = rsqrtf(r1[0] * (1.f / COUT) - mu * mu + 1e-5f);
    float* o = out + (size_t)blockIdx.x * COUT;
    for (int c = threadIdx.x; c < COUT; c += 256)
        o[c] = (p[c] - mu) * rstd * w[c] + b[c];
}

// causal depthwise conv (DCONV=4) + bias + silu; x comes from xz cols [0,DIN)
__global__ void dconv_silu_k(const float* __restrict__ xz, const float* __restrict__ w,
                             const float* __restrict__ bias, float* __restrict__ xcv) {
    int idx = blockIdx.x * 256 + threadIdx.x;            // B*L*DIN
    int d = idx & (DIN - 1);
    int row = idx >> 10, l = row & 255, b = row >> 8;
    float acc = bias[d];
    #pragma unroll
    for (int j = 0; j < 4; ++j) {
        int ll = l - 3 + j;
        if (ll >= 0) acc += xz[((size_t)(b * Lseq + ll)) * (2 * DIN) + d] * w[d * 4 + j];
    }
    xcv[(size_t)row * DIN + d] = siluf(acc);
}

// selective scan: one thread per (b, d); per-step B/C broadcast via LDS.
// xcv is read (x_t) and overwritten with the silu(z)-gated output.
__global__ __launch_bounds__(256)
void scan_k(const float* __restrict__ dbl, const float* __restrict__ dtm,
            float* __restrict__ xcv, const float* __restrict__ xz,
            const float* __restrict__ A_log, const float* __restrict__ Dskip) {
    int b = blockIdx.x >> 2;
    int d = ((blockIdx.x & 3) << 8) + threadIdx.x;
    float Ac[DSTATE];
    #pragma unroll
    for (int n = 0; n < DSTATE; ++n) Ac[n] = -__expf(A_log[(size_t)d * DSTATE + n]);
    const float Dd = Dskip[d];
    float h[DSTATE] = {};
    __shared__ float sBC[2 * DSTATE];
    for (int l = 0; l < Lseq; ++l) {
        size_t row = (size_t)b * Lseq + l;
        __syncthreads();
        if (threadIdx.x < 2 * DSTATE)
            sBC[threadIdx.x] = dbl[row * 64 + DTR + threadIdx.x];
        __syncthreads();
        float dt = dtm[row * DIN + d];
        float x  = xcv[row * DIN + d];
        float z  = xz[row * (2 * DIN) + DIN + d];
        float dtx = dt * x;
        float y = 0.f;
        #pragma unroll
        for (int n = 0; n < DSTATE; ++n) {
            float dA = __expf(dt * Ac[n]);
            h[n] = dA * h[n] + dtx * sBC[n];
            y += h[n] * sBC[DSTATE + n];
        }
        xcv[row * DIN + d] = (y + x * Dd) * siluf(z);
    }
}

// ---------------------------------------------------------------------------
extern "C" void kernel_launch(void* const* d_in, const int* in_sizes, int n_in,
                              void* d_out, int out_size, void* d_ws, size_t ws_size,
                              hipStream_t stream) {
    const float* x        = (const float*)d_in[0];
    const float* cond     = (const float*)d_in[1];
    const float* conv_w   = (const float*)d_in[2];
    const float* conv_b   = (const float*)d_in[3];
    const float* gn_w     = (const float*)d_in[4];
    const float* gn_b     = (const float*)d_in[5];
    const float* film_w   = (const float*)d_in[6];
    const float* film_b   = (const float*)d_in[7];
    const float* norm1_w  = (const float*)d_in[8];
    const float* norm1_b  = (const float*)d_in[9];
    const float* in_proj_w= (const float*)d_in[10];
    const float* dconv_w  = (const float*)d_in[11];
    const float* dconv_b  = (const float*)d_in[12];
    const float* x_proj_w = (const float*)d_in[13];
    const float* dt_proj_w= (const float*)d_in[14];
    const float* dt_proj_b= (const float*)d_in[15];
    const float* A_log    = (const float*)d_in[16];
    const float* Dskip    = (const float*)d_in[17];
    const float* out_proj_w=(const float*)d_in[18];
    const float* norm2_w  = (const float*)d_in[19];
    const float* norm2_b  = (const float*)d_in[20];
    const float* fc1_w    = (const float*)d_in[21];
    const float* fc1_b    = (const float*)d_in[22];
    const float* fc2_w    = (const float*)d_in[23];
    const float* fc2_b    = (const float*)d_in[24];
    const float* res_w    = (const float*)d_in[25];
    const float* res_b    = (const float*)d_in[26];
    float* out = (float*)d_out;

    // workspace layout (floats); buffers reused where lifetimes allow
    float* w = (float*)d_ws;
    size_t o = 0;
    float* xT       = w + o; o += (size_t)Bsz * Lseq * CIN;      // 2M
    float* Wc       = w + o; o += 768 * 512;                      // conv weight [K',N]
    float* resWT    = w + o; o += CIN * COUT;
    float* mishcond = w + o; o += Bsz * 512;
    float* embed    = w + o; o += Bsz * 2 * COUT;
    float* t1       = w + o; o += (size_t)Mrows * COUT;           // conv out -> film out
    float* ln       = w + o; o += (size_t)Mrows * COUT;           // ln1 then ln2
    float* xz       = w + o; o += (size_t)Mrows * 2 * DIN;        // xz then mlp hidden
    float* xcv      = w + o; o += (size_t)Mrows * DIN;            // conv-silu x then gated y
    float* dbl      = w + o; o += (size_t)Mrows * 64;
    float* dtm      = w + o; o += (size_t)Mrows * DIN;
    float* t2       = w + o; o += (size_t)Mrows * COUT;           // residual stream

    // --- prep ---
    transpose_x_k<<<(Bsz * Lseq * CIN) / 256, 256, 0, stream>>>(x, xT);
    prep_wc_k<<<(768 * 512) / 256, 256, 0, stream>>>(conv_w, Wc);
    prep_reswt_k<<<(CIN * COUT) / 256, 256, 0, stream>>>(res_w, resWT);
    mish_k<<<(Bsz * 512) / 256, 256, 0, stream>>>(cond, mishcond, Bsz * 512);

    // --- FiLM embed = mish(cond) @ film_w + film_b : [32, 1024]
    gemm_wmma_f32<A_NORMAL, EPI_BIAS><<<dim3(1, (2 * COUT + TNB - 1) / TNB), 256, 0, stream>>>(
        mishcond, 512, film_w, film_b, nullptr, embed, Bsz, 2 * COUT, 512);

    // --- conv1d (im2col GEMM) + bias -> t1 [B, L, COUT]
    gemm_wmma_f32<A_CONV, EPI_BIAS><<<dim3(Mrows / TM, COUT / TNB), 256, 0, stream>>>(
        xT, 0, Wc, conv_b, nullptr, t1, Mrows, COUT, 768);

    // --- groupnorm + mish + FiLM (in place)
    gn_film_k<<<Bsz * NG, 256, 0, stream>>>(t1, gn_w, gn_b, embed);

    // --- ln1
    layernorm_k<<<Mrows, 256, 0, stream>>>(t1, norm1_w, norm1_b, ln);

    // --- in_proj: xz = ln @ in_proj_w [8192, 2048]
    gemm_wmma_f32<A_NORMAL, EPI_STORE><<<dim3(Mrows / TM, (2 * DIN) / TNB), 256, 0, stream>>>(
        ln, COUT, in_proj_w, nullptr, nullptr, xz, Mrows, 2 * DIN, COUT);

    // --- causal depthwise conv + silu -> xcv
    dconv_silu_k<<<(Mrows * DIN) / 256, 256, 0, stream>>>(xz, dconv_w, dconv_b, xcv);

    // --- x_proj: dbl = xcv @ x_proj_w [8192, 64]  (N=64 < TNB, guarded)
    gemm_wmma_f32<A_NORMAL, EPI_STORE><<<dim3(Mrows / TM, 1), 256, 0, stream>>>(
        xcv, DIN, x_proj_w, nullptr, nullptr, dbl, Mrows, 64, DIN);

    // --- dt = softplus(dbl[:, :32] @ dt_proj_w + dt_proj_b) [8192, 1024]
    gemm_wmma_f32<A_NORMAL, EPI_BIAS_SOFTPLUS><<<dim3(Mrows / TM, DIN / TNB), 256, 0, stream>>>(
        dbl, 64, dt_proj_w, dt_proj_b, nullptr, dtm, Mrows, DIN, DTR);

    // --- selective scan + Dskip + silu(z) gate (xcv overwritten with y)
    scan_k<<<Bsz * (DIN / 256), 256, 0, stream>>>(dbl, dtm, xcv, xz, A_log, Dskip);

    // --- out_proj + residual t1 -> t2
    gemm_wmma_f32<A_NORMAL, EPI_ADD><<<dim3(Mrows / TM, COUT / TNB), 256, 0, stream>>>(
        xcv, DIN, out_proj_w, nullptr, t1, t2, Mrows, COUT, DIN);

    // --- ln2
    layernorm_k<<<Mrows, 256, 0, stream>>>(t2, norm2_w, norm2_b, ln);

    // --- fc1 + gelu -> hidden (reuse xz)
    gemm_wmma_f32<A_NORMAL, EPI_BIAS_GELU><<<dim3(Mrows / TM, MLPH / TNB), 256, 0, stream>>>(
        ln, COUT, fc1_w, fc1_b, nullptr, xz, Mrows, MLPH, COUT);

    // --- fc2 + bias + residual t2 -> t2 (in place, 1:1 element ownership)
    gemm_wmma_f32<A_NORMAL, EPI_BIAS_ADD><<<dim3(Mrows / TM, COUT / TNB), 256, 0, stream>>>(
        xz, MLPH, fc2_w, fc2_b, t2, t2, Mrows, COUT, MLPH);

    // --- residual 1x1 conv + bias + t2, stored transposed to [B, COUT, L]
    gemm_wmma_f32<A_NORMAL, EPI_BIAS_ADD_TRANS><<<dim3(Mrows / TM, COUT / TNB), 256, 0, stream>>>(
        xT, CIN, resWT, res_b, t2, out, Mrows, COUT, CIN);
}